// Renderer_19859928777118
// MI455X (gfx1250) — compile-verified
//
#include <hip/hip_runtime.h>

// CDNA5 / gfx1250 wave32 WMMA-based triangle rasterizer.
// Barycentric evaluation b_k(x,y) = Cx_k*x + Cy_k*y + C0_k for a 16-pixel x
// 16-face tile is computed with V_WMMA_F32_16X16X4_F32:
//   A (16x4) = [x, y, 1, 0] per pixel row, B (4x16) = [Cx; Cy; C0; 0] per face.
// A-matrix VGPR layout per ISA 7.12.2 (32-bit A 16x4): lanes 0-15 hold K=0/K=1,
// lanes 16-31 hold K=2/K=3, M = lane%16. B mirrored (lanes 0-15: K=0/1 rows,
// lanes 16-31: K=2/3 rows, N = lane%16). C/D 16x16 f32: VGPR r, lane L ->
// M = r + 8*(L>=16), N = L%16.
//
// Z-buffering runs in inverse-depth (iz) space: min depth == max iz,
// depth>NEAR <=> iz<1/NEAR, depth<FAR <=> iz>1/FAR. Only (iz, faceIdx) is
// tracked through the loop/reduction; the winner's barycentric weights are
// recomputed once per covered pixel, and the single 1/iz reciprocal matches
// the reference's depth rounding.

typedef float v2f __attribute__((ext_vector_type(2)));
typedef float v8f __attribute__((ext_vector_type(8)));

#define RIMG_H 128
#define RIMG_W 128
#define RNEAR  0.1f
#define RFAR   100.0f
#define REPS_A 1e-9f
#define IZ_LO  (1.0f / RFAR)    // 0.01  : iz must be > this  (depth < FAR)
#define IZ_HI  (1.0f / RNEAR)   // 10.0  : iz must be < this  (depth > NEAR)

// Workspace layout, BF = B*F faces:
//   float4 RecX[BF]  = (Cx0, Cx1, Cx2, 0)
//   float4 RecY[BF]  = (Cy0, Cy1, Cy2, 0)
//   float4 RecC[BF]  = (C00, C01, C02, 0)   (C00=-1 for degenerate faces)
//   float4 RecZ[BF]  = (1/z0, 1/z1, 1/z2, 0)
//   float  Tex[BF*12]  (9 used: per-vertex RGB)
// Total = 28*BF floats (~459 KB for B=2, F=2048).

__global__ __launch_bounds__(256) void face_setup_kernel(
    const float* __restrict__ vertices, const int* __restrict__ faces,
    const float* __restrict__ textures, const float* __restrict__ Km,
    const float* __restrict__ Rm, const float* __restrict__ tm,
    float* __restrict__ ws, int B, int V, int F)
{
    int idx = blockIdx.x * blockDim.x + threadIdx.x;
    int BF = B * F;
    if (idx >= BF) return;
    int b = idx / F;
    int f = idx - b * F;

    const float* Rb = Rm + b * 9;
    const float* Kb = Km + b * 9;
    const float* tb = tm + b * 3;

    int vi[3];
    vi[0] = faces[f * 3 + 0];
    vi[1] = faces[f * 3 + 1];
    vi[2] = faces[f * 3 + 2];

    float xs[3], ys[3], zs[3];
    #pragma unroll
    for (int k = 0; k < 3; ++k) {
        const float* vv = vertices + ((size_t)b * V + vi[k]) * 3;
        float v0 = vv[0], v1 = vv[1], v2 = vv[2];
        // cam_i = sum_j R[i][j] * v[j] + t[i]   (einsum 'bvj,bij->bvi')
        float cx = Rb[0]*v0 + Rb[1]*v1 + Rb[2]*v2 + tb[0];
        float cy = Rb[3]*v0 + Rb[4]*v1 + Rb[5]*v2 + tb[1];
        float cz = Rb[6]*v0 + Rb[7]*v1 + Rb[8]*v2 + tb[2];
        float pz = cz + 1e-6f;
        float px = cx / pz;
        float py = cy / pz;
        float u   = Kb[0]*px + Kb[1]*py + Kb[2];
        float vpx = Kb[3]*px + Kb[4]*py + Kb[5];
        xs[k] = u;
        ys[k] = (float)RIMG_H - vpx;
        zs[k] = cz;
    }

    float area = (xs[1]-xs[0])*(ys[2]-ys[0]) - (ys[1]-ys[0])*(xs[2]-xs[0]);
    bool  av   = fabsf(area) > REPS_A;
    float ia   = av ? (1.0f / area) : 0.0f;

    // b0 = ((x2-x1)(y-y1) - (y2-y1)(x-x1)) * ia   etc.
    float cx0 = -(ys[2]-ys[1]) * ia;
    float cy0 =  (xs[2]-xs[1]) * ia;
    float c00 = ((ys[2]-ys[1])*xs[1] - (xs[2]-xs[1])*ys[1]) * ia;
    float cx1 = -(ys[0]-ys[2]) * ia;
    float cy1 =  (xs[0]-xs[2]) * ia;
    float c01 = ((ys[0]-ys[2])*xs[2] - (xs[0]-xs[2])*ys[2]) * ia;
    float cx2 = -(ys[1]-ys[0]) * ia;
    float cy2 =  (xs[1]-xs[0]) * ia;
    float c02 = ((ys[1]-ys[0])*xs[0] - (xs[1]-xs[0])*ys[0]) * ia;

    // Degenerate faces: force b0 = -1 so the inside test always fails
    // (replaces the separate |area|>EPS flag of the reference).
    if (!av) { cx0 = 0.f; cy0 = 0.f; c00 = -1.0f; }

    float4* RecX = (float4*)ws;
    float4* RecY = RecX + BF;
    float4* RecC = RecY + BF;
    float4* RecZ = RecC + BF;
    float*  Tex  = (float*)(RecZ + BF);

    RecX[idx] = make_float4(cx0, cx1, cx2, 0.f);
    RecY[idx] = make_float4(cy0, cy1, cy2, 0.f);
    RecC[idx] = make_float4(c00, c01, c02, 0.f);
    RecZ[idx] = make_float4(1.0f/zs[0], 1.0f/zs[1], 1.0f/zs[2], 0.f);

    float* tx = Tex + (size_t)idx * 12;
    #pragma unroll
    for (int k = 0; k < 3; ++k) {
        const float* tv = textures + ((size_t)b * V + vi[k]) * 3;
        tx[k*3 + 0] = tv[0];
        tx[k*3 + 1] = tv[1];
        tx[k*3 + 2] = tv[2];
    }
    tx[9] = 0.0f; tx[10] = 0.0f; tx[11] = 0.0f;
}

__global__ __launch_bounds__(256) void raster_kernel(
    const float* __restrict__ ws, float* __restrict__ out,
    int B, int F, int totalWaves)
{
    const int HW = RIMG_H * RIMG_W;
    int lane = threadIdx.x & 31;
    int wid  = (blockIdx.x * blockDim.x + threadIdx.x) >> 5;
    if (wid >= totalWaves) return;   // whole wave exits together

    int wavesPerBatch = HW / 16;     // 1024
    int b      = wid / wavesPerBatch;
    int p_base = (wid - b * wavesPerBatch) * 16;   // 16 pixels, same image row
    int row    = p_base >> 7;        // /128
    int col0   = p_base & 127;

    int half = lane >> 4;            // 0: K=0/1 providers, 1: K=2/3 providers
    int nl   = lane & 15;            // M (A) / N (B) index

    // A matrix [x, y, 1, 0] per pixel row
    v2f a;
    a.x = half ? 1.0f : ((float)col0 + (float)nl + 0.5f);  // K0 = x  | K2 = 1
    a.y = half ? 0.0f : ((float)row + 0.5f);               // K1 = y  | K3 = 0

    int BF = B * F;
    const float4* RecX = (const float4*)ws;
    const float4* RecY = RecX + BF;
    const float4* RecC = RecY + BF;
    const float4* RecZ = RecC + BF;
    const float*  Tex  = (const float*)(RecZ + BF);

    // Lanes 0-15 feed rows K=0 (Cx) / K=1 (Cy); lanes 16-31 feed K=2 (C0) / K=3 (0).
    const float4* Prim = half ? RecC : RecX;

    float bestIz[8];
    int   bfc[8];
    #pragma unroll
    for (int r = 0; r < 8; ++r) { bestIz[r] = 0.f; bfc[r] = -1; }

    int fb = b * F;
    int nGroups = F >> 4;
    #pragma unroll 2
    for (int g = 0; g < nGroups; ++g) {
        int fi = fb + (g << 4) + nl;
        float4 cv = Prim[fi];          // Cx* (half0) or C0* (half1)
        float4 yv = RecY[fi];          // Cy* (used by half0 only)
        float4 zv = RecZ[fi];          // 1/z per vertex (all lanes)

        v2f B0, B1, B2;
        B0.x = cv.x; B0.y = half ? 0.0f : yv.x;
        B1.x = cv.y; B1.y = half ? 0.0f : yv.y;
        B2.x = cv.z; B2.y = half ? 0.0f : yv.z;

        v8f czero = {};
        v8f D0 = __builtin_amdgcn_wmma_f32_16x16x4_f32(false, a, false, B0, (short)0, czero, false, false);
        v8f D1 = __builtin_amdgcn_wmma_f32_16x16x4_f32(false, a, false, B1, (short)0, czero, false, false);
        v8f D2 = __builtin_amdgcn_wmma_f32_16x16x4_f32(false, a, false, B2, (short)0, czero, false, false);

        int faceIdx = (g << 4) + nl;
        #pragma unroll
        for (int r = 0; r < 8; ++r) {
            float e0 = D0[r], e1 = D1[r], e2 = D2[r];
            float iz = e0 * zv.x + e1 * zv.y + e2 * zv.z;
            float emin = fminf(fminf(e0, e1), e2);           // -> v_min3_num_f32
            bool valid  = (emin >= 0.f) & (iz > IZ_LO) & (iz < IZ_HI);
            float cand  = valid ? iz : 0.0f;
            bool  upd   = cand > bestIz[r];   // strict: earliest face wins ties
            bestIz[r] = upd ? cand : bestIz[r];
            bfc[r]    = upd ? faceIdx : bfc[r];
        }
    }

    // Per-pixel reduction across the 16 face-lanes of each half-wave:
    // lexicographic max over (iz, -face index) == reference scan semantics.
    float* out_rgb  = out;
    float* out_mask = out + (size_t)B * HW * 3;
    float* out_dep  = out + (size_t)B * HW * 4;

    #pragma unroll
    for (int r = 0; r < 8; ++r) {
        float iz = bestIz[r];
        int fidx = bfc[r];
        #pragma unroll
        for (int m = 8; m >= 1; m >>= 1) {   // masks 8,4,2,1 stay within 16-lane half
            float oi = __shfl_xor(iz, m, 32);
            int   of = __shfl_xor(fidx, m, 32);
            bool take = (oi > iz) || ((oi == iz) && (of >= 0) && (of < fidx));
            iz   = take ? oi : iz;
            fidx = take ? of : fidx;
        }
        if (nl == 0) {
            int pix = p_base + half * 8 + r;
            float rr = 0.f, gg = 0.f, bb = 0.f, mk = 0.f, dep = RFAR;
            if (fidx >= 0) {
                // Recompute the winner's barycentric weights (per covered pixel).
                float xpix = (float)(col0 + half * 8 + r) + 0.5f;
                float ypix = (float)row + 0.5f;
                int fo = fb + fidx;
                float4 rx = RecX[fo], ry = RecY[fo], rc = RecC[fo], rz = RecZ[fo];
                float w0 = (rx.x * xpix + ry.x * ypix + rc.x) * rz.x;
                float w1 = (rx.y * xpix + ry.y * ypix + rc.y) * rz.y;
                float w2 = (rx.z * xpix + ry.z * ypix + rc.z) * rz.z;
                float d = 1.0f / iz;     // single reciprocal per covered pixel
                const float* tx = Tex + (size_t)fo * 12;
                rr = (w0 * tx[0] + w1 * tx[3] + w2 * tx[6]) * d;
                gg = (w0 * tx[1] + w1 * tx[4] + w2 * tx[7]) * d;
                bb = (w0 * tx[2] + w1 * tx[5] + w2 * tx[8]) * d;
                mk = 1.0f;
                dep = d;
            }
            size_t gp = (size_t)b * HW + pix;
            out_rgb[gp * 3 + 0] = rr;
            out_rgb[gp * 3 + 1] = gg;
            out_rgb[gp * 3 + 2] = bb;
            out_mask[gp] = mk;
            out_dep[gp]  = dep;
        }
    }
}

extern "C" void kernel_launch(void* const* d_in, const int* in_sizes, int n_in,
                              void* d_out, int out_size, void* d_ws, size_t ws_size,
                              hipStream_t stream)
{
    (void)n_in; (void)out_size; (void)ws_size;
    const float* vertices = (const float*)d_in[0];
    const int*   faces    = (const int*)d_in[1];
    const float* textures = (const float*)d_in[2];
    const float* K        = (const float*)d_in[3];
    const float* R        = (const float*)d_in[4];
    const float* t        = (const float*)d_in[5];

    int B = in_sizes[3] / 9;          // K is (B,3,3)
    int F = in_sizes[1] / 3;          // faces is (F,3)
    int V = in_sizes[0] / (3 * B);    // vertices is (B,V,3)

    float* ws  = (float*)d_ws;        // needs 28*B*F floats (~459 KB for B=2,F=2048)
    float* out = (float*)d_out;

    int BF = B * F;
    face_setup_kernel<<<(BF + 255) / 256, 256, 0, stream>>>(
        vertices, faces, textures, K, R, t, ws, B, V, F);

    int totalWaves = B * ((RIMG_H * RIMG_W) / 16);
    int threads    = totalWaves * 32;
    raster_kernel<<<(threads + 255) / 256, 256, 0, stream>>>(ws, out, B, F, totalWaves);
}